// ExpertLayer_65644280152196
// MI455X (gfx1250) — compile-verified
//
#include <hip/hip_runtime.h>
#include <cstdint>
#include <cstddef>

// Problem sizes (from reference): B=2, S=1024, D=512, H=2048, E=64, K=2
#define BB   2
#define SS   1024
#define DD   512
#define HH   2048
#define EE   64
#define KKK  2
#define NTOK (BB * SS)          // 2048 tokens
#define LNEPS 1e-5f

typedef float v2f __attribute__((ext_vector_type(2)));
typedef float v8f __attribute__((ext_vector_type(8)));

// CDNA5 fp32 WMMA: D(16x16,f32) = A(16x4,f32) * B(4x16,f32) + C
static __device__ __forceinline__ v8f wmma_f32(v2f a, v2f b, v8f c) {
  return __builtin_amdgcn_wmma_f32_16x16x4_f32(
      /*neg_a=*/false, a, /*neg_b=*/false, b,
      /*c_mod=*/(short)0, c, /*reuse_a=*/false, /*reuse_b=*/false);
}

// ---------------------------------------------------------------------------
// K0: zero the output accumulation buffer (d_out) + expert counters.
// ---------------------------------------------------------------------------
__global__ __launch_bounds__(256) void zero_kernel(float4* __restrict__ out4,
                                                   int* __restrict__ counts,
                                                   int* __restrict__ cursors) {
  const int i = blockIdx.x * 256 + threadIdx.x;
  out4[i] = make_float4(0.f, 0.f, 0.f, 0.f);
  if (blockIdx.x == 0 && threadIdx.x < EE) {
    counts[threadIdx.x]  = 0;
    cursors[threadIdx.x] = 0;
  }
}

// ---------------------------------------------------------------------------
// K1: gating. One 64-thread block per token: logits = x@Wg + bg, softmax,
// top-2 (first-occurrence ties, like jax top_k), histogram expert counts.
// ---------------------------------------------------------------------------
__global__ __launch_bounds__(64) void gating_kernel(
    const float* __restrict__ x, const float* __restrict__ Wg,
    const float* __restrict__ bg, int* __restrict__ topk_idx,
    float* __restrict__ topk_w, int* __restrict__ counts) {
  __shared__ float xs[DD];
  __shared__ float lg[EE];
  const int t = blockIdx.x;
  const int tid = threadIdx.x;

  for (int i = tid; i < DD; i += 64) xs[i] = x[(size_t)t * DD + i];
  __syncthreads();

  float acc = bg[tid];
#pragma unroll 8
  for (int d = 0; d < DD; ++d) acc += xs[d] * Wg[(size_t)d * EE + tid];
  lg[tid] = acc;
  __syncthreads();

  if (tid == 0) {
    float maxv = lg[0];
    for (int i = 1; i < EE; ++i) maxv = lg[i] > maxv ? lg[i] : maxv;
    float sum = 0.f;
    for (int i = 0; i < EE; ++i) sum += expf(lg[i] - maxv);
    // top-1
    int i1 = 0; float v1 = lg[0];
    for (int i = 1; i < EE; ++i) if (lg[i] > v1) { v1 = lg[i]; i1 = i; }
    // top-2
    int i2 = (i1 == 0) ? 1 : 0; float v2 = lg[i2];
    for (int i = 0; i < EE; ++i)
      if (i != i1 && lg[i] > v2) { v2 = lg[i]; i2 = i; }

    const float inv = 1.f / sum;
    topk_idx[t * 2 + 0] = i1;
    topk_idx[t * 2 + 1] = i2;
    topk_w[t * 2 + 0] = expf(v1 - maxv) * inv;
    topk_w[t * 2 + 1] = expf(v2 - maxv) * inv;
    atomicAdd(&counts[i1], 1);
    atomicAdd(&counts[i2], 1);
  }
}

// ---------------------------------------------------------------------------
// K2: exclusive scan of expert counts (E=64, trivial serial).
// ---------------------------------------------------------------------------
__global__ void scan_kernel(const int* __restrict__ counts,
                            int* __restrict__ offsets) {
  if (threadIdx.x == 0) {
    int acc = 0;
    for (int e = 0; e < EE; ++e) { offsets[e] = acc; acc += counts[e]; }
    offsets[EE] = acc;
  }
}

// ---------------------------------------------------------------------------
// K3: scatter tokens into per-expert contiguous lists.
// ---------------------------------------------------------------------------
__global__ __launch_bounds__(256) void scatter_kernel(
    const int* __restrict__ topk_idx, const float* __restrict__ topk_w,
    const int* __restrict__ offsets, int* __restrict__ cursors,
    int* __restrict__ token_list, float* __restrict__ weight_list) {
  const int t = blockIdx.x * 256 + threadIdx.x;
  if (t >= NTOK) return;
#pragma unroll
  for (int k = 0; k < KKK; ++k) {
    const int e = topk_idx[t * 2 + k];
    const int p = atomicAdd(&cursors[e], 1);
    const int j = offsets[e] + p;
    token_list[j]  = t;
    weight_list[j] = topk_w[t * 2 + k];
  }
}

// ---------------------------------------------------------------------------
// K4: grouped expert GEMM. Block = (expert e, tile of 16 tokens), 8 wave32s.
//   stage 1: H[16x2048] = relu(X[16x512] @ W1[e] + b1[e])  (WMMA f32 16x16x4)
//   stage 2: O[16x512]  = H @ W2[e] + b2[e]; atomic-add gate_w * O into out.
// LDS: Xs 16x516 (33 KB) + Hs 16x2052 (128 KB), padded strides to kill
// bank conflicts on A-fragment reads.
// ---------------------------------------------------------------------------
#define XSTR 516
#define HSTR 2052

__global__ __launch_bounds__(256) void expert_kernel(
    const float* __restrict__ x, const float* __restrict__ W1,
    const float* __restrict__ b1, const float* __restrict__ W2,
    const float* __restrict__ b2, const int* __restrict__ offsets,
    const int* __restrict__ token_list, const float* __restrict__ weight_list,
    float* __restrict__ out) {
  __shared__ __align__(16) float Xs[16 * XSTR];
  __shared__ __align__(16) float Hs[16 * HSTR];
  __shared__ int   stok[16];
  __shared__ float sgw[16];

  const int e     = blockIdx.x;
  const int base  = offsets[e];
  const int nt    = offsets[e + 1] - base;
  const int tile0 = blockIdx.y * 16;
  if (tile0 >= nt) return;                 // uniform early-exit for empty tiles

  const int tid  = threadIdx.x;
  const int wave = tid >> 5;
  const int lane = tid & 31;
  const int half = lane >> 4;              // ISA A/B layout: lanes 16-31 hold K+2
  const int m    = lane & 15;

  if (tid < 16) {
    const int g = tile0 + tid;
    if (g < nt) { stok[tid] = token_list[base + g]; sgw[tid] = weight_list[base + g]; }
    else        { stok[tid] = -1; sgw[tid] = 0.f; }
  }
  __syncthreads();

  for (int i = tid; i < 16 * DD; i += 256) {
    const int s = i >> 9, d = i & (DD - 1);
    const int tk = stok[s];
    Xs[s * XSTR + d] = (tk >= 0) ? x[(size_t)tk * DD + d] : 0.f;
  }
  __syncthreads();

  const float* W1e = W1 + (size_t)e * DD * HH;
  const float* b1e = b1 + (size_t)e * HH;
  const float* W2e = W2 + (size_t)e * HH * DD;
  const float* b2e = b2 + (size_t)e * DD;

  // ---------------- stage 1: this wave owns H columns [c1, c1+256) ---------
  const int c1 = wave * 256;
  v8f zero = {0.f, 0.f, 0.f, 0.f, 0.f, 0.f, 0.f, 0.f};
  v8f acc[16];
#pragma unroll
  for (int t = 0; t < 16; ++t) acc[t] = zero;

  for (int k = 0; k < DD; k += 4) {
    const v2f a = *(const v2f*)&Xs[m * XSTR + k + 2 * half];
    const float* Bp = W1e + (size_t)(k + 2 * half) * HH + c1 + m;
    __builtin_prefetch(Bp + 4 * (size_t)HH, 0, 1);   // next K-panel of W1
#pragma unroll
    for (int t = 0; t < 16; ++t) {
      v2f b; b.x = Bp[t * 16]; b.y = Bp[t * 16 + HH];
      acc[t] = wmma_f32(a, b, acc[t]);
    }
  }

#pragma unroll
  for (int t = 0; t < 16; ++t) {
    const int col = c1 + t * 16 + m;
    const float bias = b1e[col];
#pragma unroll
    for (int r = 0; r < 8; ++r) {
      float v = acc[t][r] + bias;           // C layout: M = r + 8*half
      v = v > 0.f ? v : 0.f;                // ReLU
      Hs[(r + 8 * half) * HSTR + col] = v;
    }
  }
  __syncthreads();

  // ---------------- stage 2: this wave owns D columns [c2, c2+64) ----------
  const int c2 = wave * 64;
  v8f o[4];
#pragma unroll
  for (int t = 0; t < 4; ++t) o[t] = zero;

  for (int k = 0; k < HH; k += 4) {
    const v2f a = *(const v2f*)&Hs[m * HSTR + k + 2 * half];
    const float* Bp = W2e + (size_t)(k + 2 * half) * DD + c2 + m;
#pragma unroll
    for (int t = 0; t < 4; ++t) {
      v2f b; b.x = Bp[t * 16]; b.y = Bp[t * 16 + DD];
      o[t] = wmma_f32(a, b, o[t]);
    }
  }

#pragma unroll
  for (int t = 0; t < 4; ++t) {
    const int col = c2 + t * 16 + m;
    const float bias = b2e[col];
#pragma unroll
    for (int r = 0; r < 8; ++r) {
      const int s  = r + 8 * half;
      const int tk = stok[s];
      if (tk >= 0)
        atomicAdd(&out[(size_t)tk * DD + col], (o[t][r] + bias) * sgw[s]);
    }
  }
}

// ---------------------------------------------------------------------------
// K5: residual + LayerNorm, in place on d_out. One block per token.
// ---------------------------------------------------------------------------
__global__ __launch_bounds__(256) void finalize_kernel(
    float* __restrict__ out, const float* __restrict__ x,
    const float* __restrict__ gamma, const float* __restrict__ beta) {
  __shared__ float rs[256];
  __shared__ float rq[256];
  const int t = blockIdx.x, tid = threadIdx.x;
  float* row = out + (size_t)t * DD;
  const float* xr = x + (size_t)t * DD;

  const float o0 = row[tid]       + xr[tid];
  const float o1 = row[tid + 256] + xr[tid + 256];
  rs[tid] = o0 + o1;
  rq[tid] = o0 * o0 + o1 * o1;
  __syncthreads();
  for (int off = 128; off > 0; off >>= 1) {
    if (tid < off) { rs[tid] += rs[tid + off]; rq[tid] += rq[tid + off]; }
    __syncthreads();
  }
  const float mu   = rs[0] * (1.f / DD);
  const float var  = rq[0] * (1.f / DD) - mu * mu;
  const float rstd = rsqrtf(var + LNEPS);
  row[tid]       = (o0 - mu) * rstd * gamma[tid]       + beta[tid];
  row[tid + 256] = (o1 - mu) * rstd * gamma[tid + 256] + beta[tid + 256];
}

// ---------------------------------------------------------------------------
extern "C" void kernel_launch(void* const* d_in, const int* in_sizes, int n_in,
                              void* d_out, int out_size, void* d_ws, size_t ws_size,
                              hipStream_t stream) {
  const float* x     = (const float*)d_in[0];
  const float* Wg    = (const float*)d_in[1];
  const float* bg    = (const float*)d_in[2];
  const float* W1    = (const float*)d_in[3];
  const float* b1    = (const float*)d_in[4];
  const float* W2    = (const float*)d_in[5];
  const float* b2    = (const float*)d_in[6];
  const float* gamma = (const float*)d_in[7];
  const float* beta  = (const float*)d_in[8];
  float* out = (float*)d_out;
  (void)in_sizes; (void)n_in; (void)out_size; (void)ws_size;

  uint8_t* w = (uint8_t*)d_ws;
  int*   topk_idx    = (int*)w;   w += (size_t)NTOK * KKK * sizeof(int);
  float* topk_w      = (float*)w; w += (size_t)NTOK * KKK * sizeof(float);
  int*   counts      = (int*)w;   w += 256;
  int*   offsets     = (int*)w;   w += 512;   // E+1 = 65 ints
  int*   cursors     = (int*)w;   w += 256;
  int*   token_list  = (int*)w;   w += (size_t)NTOK * KKK * sizeof(int);
  float* weight_list = (float*)w;

  zero_kernel<<<dim3((NTOK * DD) / (4 * 256)), dim3(256), 0, stream>>>(
      (float4*)out, counts, cursors);
  gating_kernel<<<dim3(NTOK), dim3(64), 0, stream>>>(
      x, Wg, bg, topk_idx, topk_w, counts);
  scan_kernel<<<dim3(1), dim3(32), 0, stream>>>(counts, offsets);
  scatter_kernel<<<dim3((NTOK + 255) / 256), dim3(256), 0, stream>>>(
      topk_idx, topk_w, offsets, cursors, token_list, weight_list);
  // worst case one expert owns all 2048 tokens -> 128 tiles; empty tiles exit.
  expert_kernel<<<dim3(EE, 128), dim3(256), 0, stream>>>(
      x, W1, b1, W2, b2, offsets, token_list, weight_list, out);
  finalize_kernel<<<dim3(NTOK), dim3(256), 0, stream>>>(out, x, gamma, beta);
}